// InstrumentSpecificSlotMemory_82978768159042
// MI455X (gfx1250) — compile-verified
//
#include <hip/hip_runtime.h>
#include <math.h>

#define DD 768
#define KK 17
#define BB 16
#define TT 4096
#define LL 1024
#define TTILE 256
#define NTB (TT / TTILE) /* 16 */
#define APAD 772         /* 768 + 4: LDS row stride for manual staging */
#define DSTRIDE 780      /* 768 + 3*4: row stride produced by TDM padding */

typedef float v2f __attribute__((ext_vector_type(2)));
typedef float v8f __attribute__((ext_vector_type(8)));
typedef uint32_t v4u __attribute__((ext_vector_type(4)));
typedef int v8i __attribute__((ext_vector_type(8)));

__device__ __forceinline__ v8f wmma4(v2f a, v2f b, v8f c) {
  // D = A(16x4 f32) * B(4x16 f32) + C(16x16 f32)
  return __builtin_amdgcn_wmma_f32_16x16x4_f32(false, a, false, b, (short)0, c,
                                               false, false);
}

__device__ __forceinline__ float sig_(float x) {
  return 1.0f / (1.0f + __expf(-x));
}

__device__ __forceinline__ float wred(float s) {
#pragma unroll
  for (int off = 16; off; off >>= 1) s += __shfl_xor(s, off, 32);
  return s;
}

// LDS byte offset of a generic pointer that aliases __shared__ memory.
__device__ __forceinline__ uint32_t lds_addr(const void* p) {
  return (uint32_t)(uintptr_t)(__attribute__((address_space(3)))
                               const uint8_t*)(const uint8_t*)p;
}

// Async copy: n4 16-byte chunks, global-contiguous -> LDS-contiguous.
__device__ __forceinline__ void async_copy_contig(const float* __restrict__ g,
                                                  const float* l, int n4,
                                                  int tid) {
  const uint32_t base = lds_addr(l);
  for (int i = tid; i < n4; i += 256) {
    const uint64_t ga = (uint64_t)(uintptr_t)(g + (size_t)i * 4);
    const uint32_t la = base + (uint32_t)i * 16u;
    asm volatile("global_load_async_to_lds_b128 %0, %1, off" ::"v"(la), "v"(ga)
                 : "memory");
  }
  asm volatile("s_wait_asynccnt 0x0" ::: "memory");
}

// Async copy of a 16 x DD f32 tile (global row stride DD) into LDS with row
// stride APAD (per-lane LDS destination addresses make the padding free).
__device__ __forceinline__ void async_copy_tile(const float* __restrict__ g,
                                                const float* l, int tid) {
  const uint32_t base = lds_addr(l);
  for (int i = tid; i < 16 * (DD / 4); i += 256) {
    const int r = i / (DD / 4);
    const int c4 = i - r * (DD / 4);
    const uint64_t ga = (uint64_t)(uintptr_t)(g + (size_t)r * DD + c4 * 4);
    const uint32_t la = base + (uint32_t)(r * APAD + c4 * 4) * 4u;
    asm volatile("global_load_async_to_lds_b128 %0, %1, off" ::"v"(la), "v"(ga)
                 : "memory");
  }
  asm volatile("s_wait_asynccnt 0x0" ::: "memory");
}

// Column position inside a TDM-padded row (4 pad DWORDs every 256 DWORDs).
__device__ __forceinline__ int dpos(int c) { return c + ((c >> 8) << 2); }

// ---------------------------------------------------------------------------
// Kernel 1: fused routing + aggregation partials.
// Phase A: route = sigmoid(H_tile @ Wr^T + br)  (Wr in LDS via async DMA)
// Phase B: partial_agg(17x768) = route^T @ H_tile (deterministic per-block)
// ---------------------------------------------------------------------------
extern "C" __global__ void __launch_bounds__(256)
k_route_agg(const float* __restrict__ H, const float* __restrict__ Wr,
            const float* __restrict__ br, float* __restrict__ partial) {
  extern __shared__ float sm[];
  float* wr = sm;                  // 17*768
  float* route = wr + KK * DD;     // 256*17
  float* brl = route + TTILE * KK; // 17
  const int tb = blockIdx.x, b = blockIdx.y;
  const int tid = threadIdx.x, lane = tid & 31, wid = tid >> 5;
  const int t0 = tb * TTILE;

  async_copy_contig(Wr, wr, KK * DD / 4, tid); // 17*768 contiguous
  if (tid < KK) brl[tid] = br[tid];
  __syncthreads();

  // Phase A: each wave handles 32 rows
  for (int i = 0; i < 32; ++i) {
    const int t = wid * 32 + i;
    const float* hrow = H + ((size_t)b * TT + t0 + t) * DD;
    float h[24];
#pragma unroll
    for (int j = 0; j < 24; ++j) h[j] = hrow[lane + 32 * j];
    for (int k = 0; k < KK; ++k) {
      float s = 0.f;
#pragma unroll
      for (int j = 0; j < 24; ++j) s = fmaf(h[j], wr[k * DD + lane + 32 * j], s);
      s = wred(s);
      if (lane == 0) route[t * KK + k] = sig_(s + brl[k]);
    }
  }
  __syncthreads();

  // Phase B: thread owns 3 d-columns, loops all 256 rows (L2-warm H reload)
  float a0[KK], a1[KK], a2[KK];
#pragma unroll
  for (int k = 0; k < KK; ++k) { a0[k] = 0.f; a1[k] = 0.f; a2[k] = 0.f; }
  const int d0 = tid;
  for (int t = 0; t < TTILE; ++t) {
    const float* hrow = H + ((size_t)b * TT + t0 + t) * DD;
    const float h0 = hrow[d0], h1 = hrow[d0 + 256], h2 = hrow[d0 + 512];
#pragma unroll
    for (int k = 0; k < KK; ++k) {
      const float r = route[t * KK + k];
      a0[k] = fmaf(r, h0, a0[k]);
      a1[k] = fmaf(r, h1, a1[k]);
      a2[k] = fmaf(r, h2, a2[k]);
    }
  }
  float* p = partial + ((size_t)(b * NTB + tb)) * KK * DD;
#pragma unroll
  for (int k = 0; k < KK; ++k) {
    p[k * DD + d0] = a0[k];
    p[k * DD + d0 + 256] = a1[k];
    p[k * DD + d0 + 512] = a2[k];
  }
}

// ---------------------------------------------------------------------------
// Kernel 2: agg[b,k,d] = sum over 16 T-tiles of partials (deterministic order)
// ---------------------------------------------------------------------------
extern "C" __global__ void __launch_bounds__(256)
k_reduce(const float* __restrict__ partial, float* __restrict__ agg) {
  const int idx = blockIdx.x * 256 + threadIdx.x; // 16*17*768 = 208896
  if (idx >= BB * KK * DD) return;
  const int b = idx / (KK * DD);
  const int rem = idx - b * KK * DD;
  const float* p = partial + (size_t)b * NTB * KK * DD + rem;
  float s = 0.f;
#pragma unroll
  for (int tb = 0; tb < NTB; ++tb) s += p[(size_t)tb * KK * DD];
  agg[idx] = s;
}

// ---------------------------------------------------------------------------
// Kernel 3/5: O = A(272xD) @ W^T(DxN) via V_WMMA_F32_16X16X4_F32.
// A m-tile staged by async-to-LDS DMA (padded stride); B streamed from
// L2-resident W. blockIdx.z picks (A,W,O) pair: GI/GH or Kt/V.
// ---------------------------------------------------------------------------
extern "C" __global__ void __launch_bounds__(256)
k_gemm_awt(const float* __restrict__ A0, const float* __restrict__ A1,
           const float* __restrict__ W0, const float* __restrict__ W1,
           float* __restrict__ O0, float* __restrict__ O1, int N) {
  __shared__ float As[16 * APAD];
  const float* Ain = blockIdx.z ? A1 : A0;
  const float* Win = blockIdx.z ? W1 : W0;
  float* Out = blockIdx.z ? O1 : O0;
  const int mt = blockIdx.y;
  const int tid = threadIdx.x, lane = tid & 31, wid = tid >> 5;

  async_copy_tile(Ain + (size_t)mt * 16 * DD, As, tid);
  __syncthreads();

  const int n0 = (blockIdx.x * 8 + wid) * 16;
  const int m = lane & 15, hi = lane >> 4;
  const float* wrow = Win + (size_t)(n0 + m) * DD; // B[k][n] = W[n][k]
  v8f acc = {0.f, 0.f, 0.f, 0.f, 0.f, 0.f, 0.f, 0.f};
  for (int kk = 0; kk < DD / 4; ++kk) {
    const int koff = 4 * kk + 2 * hi;
    v2f a, bf;
    a.x = As[m * APAD + koff];
    a.y = As[m * APAD + koff + 1];
    bf.x = wrow[koff];
    bf.y = wrow[koff + 1];
    acc = wmma4(a, bf, acc);
  }
#pragma unroll
  for (int r = 0; r < 8; ++r) {
    const int mg = mt * 16 + r + 8 * hi;
    Out[(size_t)mg * N + n0 + m] = acc[r];
  }
}

// ---------------------------------------------------------------------------
// Kernel 4: elementwise GRU (r,z,n order) + LayerNorm -> M_new
// ---------------------------------------------------------------------------
extern "C" __global__ void __launch_bounds__(256)
k_gru_ln(const float* __restrict__ GI, const float* __restrict__ GH,
         const float* __restrict__ Mold, const float* __restrict__ b_ih,
         const float* __restrict__ b_hh, const float* __restrict__ g,
         const float* __restrict__ be, float* __restrict__ Mnew) {
  __shared__ float red[256];
  __shared__ float stats[2];
  const int m = blockIdx.x, tid = threadIdx.x;
  float h[3];
#pragma unroll
  for (int jj = 0; jj < 3; ++jj) {
    const int d = tid + jj * 256;
    const size_t base = (size_t)m * 3 * DD;
    const float gir = GI[base + d] + b_ih[d];
    const float giz = GI[base + DD + d] + b_ih[DD + d];
    const float gin = GI[base + 2 * DD + d] + b_ih[2 * DD + d];
    const float ghr = GH[base + d] + b_hh[d];
    const float ghz = GH[base + DD + d] + b_hh[DD + d];
    const float ghn = GH[base + 2 * DD + d] + b_hh[2 * DD + d];
    const float r = sig_(gir + ghr);
    const float z = sig_(giz + ghz);
    const float n = tanhf(gin + r * ghn);
    h[jj] = (1.f - z) * n + z * Mold[(size_t)m * DD + d];
  }
  red[tid] = h[0] + h[1] + h[2];
  __syncthreads();
  for (int o = 128; o; o >>= 1) {
    if (tid < o) red[tid] += red[tid + o];
    __syncthreads();
  }
  if (tid == 0) stats[0] = red[0] * (1.f / DD);
  __syncthreads();
  const float mean = stats[0];
  float q = 0.f;
#pragma unroll
  for (int jj = 0; jj < 3; ++jj) {
    const float dm = h[jj] - mean;
    q += dm * dm;
  }
  __syncthreads();
  red[tid] = q;
  __syncthreads();
  for (int o = 128; o; o >>= 1) {
    if (tid < o) red[tid] += red[tid + o];
    __syncthreads();
  }
  if (tid == 0) stats[1] = rsqrtf(red[0] * (1.f / DD) + 1e-5f);
  __syncthreads();
  const float rstd = stats[1];
#pragma unroll
  for (int jj = 0; jj < 3; ++jj) {
    const int d = tid + jj * 256;
    Mnew[(size_t)m * DD + d] = (h[jj] - mean) * rstd * g[d] + be[d];
  }
}

// ---------------------------------------------------------------------------
// Kernel 6: fused attention. dec tile loaded by the Tensor Data Mover with
// hardware row padding (4 DWORDs / 256 DWORDs -> row stride 780, bank-safe).
// Q = dec_tile @ Wq^T via WMMA, then 17-wide softmax/ctx/LN/residual.
// ---------------------------------------------------------------------------
extern "C" __global__ void __launch_bounds__(256)
k_attn(const float* __restrict__ dec, const float* __restrict__ Wq,
       const float* __restrict__ Kt, const float* __restrict__ V,
       const float* __restrict__ g, const float* __restrict__ be,
       const float* __restrict__ mgate, float* __restrict__ out) {
  extern __shared__ float sm[];
  float* decs = sm;               // 16*DSTRIDE (TDM-padded)
  float* qs = sm + 16 * DSTRIDE;  // 16*APAD
  const int lt = blockIdx.x, b = blockIdx.y;
  const int tid = threadIdx.x, lane = tid & 31, wid = tid >> 5;
  const int l0 = lt * 16;

  const float* drow = dec + ((size_t)b * LL + l0) * DD;
  if (wid == 0) {
    // Tensor DMA descriptor: 2D tile 768x16 f32, LDS padding enabled.
    const uint64_t ga = (uint64_t)(uintptr_t)drow;
    v4u g0;
    g0[0] = 1u;               // count=1, user descriptor
    g0[1] = lds_addr(decs);   // lds_addr
    g0[2] = (uint32_t)ga;     // global_addr[31:0]
    g0[3] = (uint32_t)((ga >> 32) & 0x01FFFFFFu) | 0x80000000u; // type=2
    v8i g1;
    g1[0] = (int)0x07D20000;  // data_size=4B, pad_en, interval=256dw, amt=4dw
    g1[1] = (int)(768u << 16);        // tensor_dim0 = 768
    g1[2] = (int)(16u << 16);         // tensor_dim1 = 16
    g1[3] = (int)(768u << 16);        // tile_dim0 = 768
    g1[4] = 16;                       // tile_dim1 = 16, tile_dim2 = 0
    g1[5] = 768;                      // tensor_dim0_stride = 768
    g1[6] = 0;
    g1[7] = 0;
    asm volatile("tensor_load_to_lds %0, %1" ::"s"(g0), "s"(g1) : "memory");
    __builtin_amdgcn_s_wait_tensorcnt(0);
  }
  __syncthreads();

  // Q projection: wave wid owns N-tiles [wid*6, wid*6+6)
  const int m = lane & 15, hi = lane >> 4;
  v8f acc[6];
#pragma unroll
  for (int j = 0; j < 6; ++j)
    acc[j] = (v8f){0.f, 0.f, 0.f, 0.f, 0.f, 0.f, 0.f, 0.f};
  for (int kk = 0; kk < DD / 4; ++kk) {
    const int koff = 4 * kk + 2 * hi; // even: never crosses a pad boundary
    const int kp = dpos(koff);
    v2f a;
    a.x = decs[m * DSTRIDE + kp];
    a.y = decs[m * DSTRIDE + kp + 1];
#pragma unroll
    for (int j = 0; j < 6; ++j) {
      const int n0 = (wid * 6 + j) * 16;
      const float* wrow = Wq + (size_t)(n0 + m) * DD;
      v2f bf;
      bf.x = wrow[koff];
      bf.y = wrow[koff + 1];
      acc[j] = wmma4(a, bf, acc[j]);
    }
  }
#pragma unroll
  for (int j = 0; j < 6; ++j) {
    const int n0 = (wid * 6 + j) * 16;
#pragma unroll
    for (int r = 0; r < 8; ++r) qs[(r + 8 * hi) * APAD + n0 + m] = acc[j][r];
  }
  __syncthreads();

  const float gate = mgate[0];
  const float scale = 0.03608439182435161f; // 1/sqrt(768)
  for (int half = 0; half < 2; ++half) {
    const int mm = wid + 8 * half;
    float qreg[24];
#pragma unroll
    for (int j = 0; j < 24; ++j) qreg[j] = qs[mm * APAD + lane + 32 * j];
    float sc[KK];
#pragma unroll
    for (int k = 0; k < KK; ++k) {
      const float* krow = Kt + ((size_t)(b * KK + k)) * DD;
      float s = 0.f;
#pragma unroll
      for (int j = 0; j < 24; ++j) s = fmaf(qreg[j], krow[lane + 32 * j], s);
      sc[k] = wred(s) * scale;
    }
    float mx = sc[0];
#pragma unroll
    for (int k = 1; k < KK; ++k) mx = fmaxf(mx, sc[k]);
    float se = 0.f;
#pragma unroll
    for (int k = 0; k < KK; ++k) {
      sc[k] = __expf(sc[k] - mx);
      se += sc[k];
    }
    const float inv = 1.f / se;
    float cx[24];
#pragma unroll
    for (int j = 0; j < 24; ++j) cx[j] = 0.f;
#pragma unroll
    for (int k = 0; k < KK; ++k) {
      const float* vrow = V + ((size_t)(b * KK + k)) * DD;
      const float aw = sc[k] * inv;
#pragma unroll
      for (int j = 0; j < 24; ++j) cx[j] = fmaf(aw, vrow[lane + 32 * j], cx[j]);
    }
    float s1 = 0.f;
#pragma unroll
    for (int j = 0; j < 24; ++j) s1 += cx[j];
    const float mean = wred(s1) * (1.f / DD);
    float s2 = 0.f;
#pragma unroll
    for (int j = 0; j < 24; ++j) {
      const float dm = cx[j] - mean;
      s2 += dm * dm;
    }
    const float rstd = rsqrtf(wred(s2) * (1.f / DD) + 1e-5f);
    float* orow = out + ((size_t)b * LL + l0 + mm) * DD;
#pragma unroll
    for (int j = 0; j < 24; ++j) {
      const int d = lane + 32 * j;
      orow[d] = decs[mm * DSTRIDE + dpos(d)] +
                gate * ((cx[j] - mean) * rstd * g[d] + be[d]);
    }
  }
}

// ---------------------------------------------------------------------------
extern "C" void kernel_launch(void* const* d_in, const int* in_sizes, int n_in,
                              void* d_out, int out_size, void* d_ws,
                              size_t ws_size, hipStream_t stream) {
  (void)in_sizes; (void)n_in; (void)out_size; (void)ws_size;
  const float* H = (const float*)d_in[0];
  const float* Mold = (const float*)d_in[1];
  const float* dec = (const float*)d_in[2];
  const float* Wr = (const float*)d_in[3];
  const float* br = (const float*)d_in[4];
  const float* W_ih = (const float*)d_in[5];
  const float* W_hh = (const float*)d_in[6];
  const float* b_ih = (const float*)d_in[7];
  const float* b_hh = (const float*)d_in[8];
  const float* g_mem = (const float*)d_in[9];
  const float* b_mem = (const float*)d_in[10];
  const float* Wq = (const float*)d_in[11];
  const float* Wk = (const float*)d_in[12];
  const float* Wv = (const float*)d_in[13];
  const float* g_at = (const float*)d_in[14];
  const float* b_at = (const float*)d_in[15];
  const float* mg = (const float*)d_in[16];
  float* out = (float*)d_out;

  float* ws = (float*)d_ws;
  float* partial = ws;                               // 256*17*768
  float* agg = partial + (size_t)BB * NTB * KK * DD; // 16*17*768
  float* GI = agg + (size_t)BB * KK * DD;            // 272*2304
  float* GH = GI + (size_t)BB * KK * 3 * DD;         // 272*2304
  float* Mnew = GH + (size_t)BB * KK * 3 * DD;       // 272*768
  float* Ktb = Mnew + (size_t)BB * KK * DD;          // 272*768
  float* Vb = Ktb + (size_t)BB * KK * DD;            // 272*768

  const size_t lds1 = (size_t)(KK * DD + TTILE * KK + 32) * sizeof(float);
  k_route_agg<<<dim3(NTB, BB), 256, lds1, stream>>>(H, Wr, br, partial);
  k_reduce<<<dim3((BB * KK * DD + 255) / 256), 256, 0, stream>>>(partial, agg);
  // gates: N = 2304 -> 144 N-tiles -> grid.x = 18; z: (agg,W_ih)/(M_old,W_hh)
  k_gemm_awt<<<dim3(18, 17, 2), 256, 0, stream>>>(agg, Mold, W_ih, W_hh, GI, GH,
                                                  3 * DD);
  k_gru_ln<<<dim3(BB * KK), 256, 0, stream>>>(GI, GH, Mold, b_ih, b_hh, g_mem,
                                              b_mem, Mnew);
  // Kt/V: N = 768 -> 48 N-tiles -> grid.x = 6; z picks Wk/Wv
  k_gemm_awt<<<dim3(6, 17, 2), 256, 0, stream>>>(Mnew, Mnew, Wk, Wv, Ktb, Vb,
                                                 DD);
  const size_t lds6 = (size_t)(16 * DSTRIDE + 16 * APAD) * sizeof(float);
  k_attn<<<dim3(LL / 16, BB), 256, lds6, stream>>>(dec, Wq, Ktb, Vb, g_at, b_at,
                                                   mg, out);
}